// GraphAugmentation_61512521613372
// MI455X (gfx1250) — compile-verified
//
#include <hip/hip_runtime.h>
#include <hip/hip_bf16.h>
#include <math.h>
#include <stdint.h>

// Shapes: B=16, C=D=16, H=W=256.
#define Hh 256
#define Ww 256
#define Cc 16
#define Bb 16

typedef float v2f __attribute__((ext_vector_type(2)));
typedef float v8f __attribute__((ext_vector_type(8)));

struct DyInfo {
    int dy[8];      // dy per offset k (dx is a no-op in the reference's _shift_pad)
    int jmap[8];    // k -> distinct-dy slot
    int dylist[8];  // distinct dy values
    int ndy;
};

// ---------------------------------------------------------------------------
// Kernel 1: row sums of x: rowsum[(b*16+c)*256 + r] = sum_x x[b,c,r,:]
// one wave32 per row, float4 loads, xor-shuffle reduction
// ---------------------------------------------------------------------------
__global__ __launch_bounds__(256) void k1_rowsum(const float* __restrict__ x,
                                                 float* __restrict__ rowsum) {
    int t = threadIdx.x;
    int wave = t >> 5, lane = t & 31;
    int row = blockIdx.x * 8 + wave;                 // 0 .. 65535
    const float4* p = (const float4*)(x + (size_t)row * Ww);
    float4 v0 = p[lane], v1 = p[lane + 32];
    float s = v0.x + v0.y + v0.z + v0.w + v1.x + v1.y + v1.z + v1.w;
    #pragma unroll
    for (int off = 16; off >= 1; off >>= 1) s += __shfl_xor(s, off, 32);
    if (lane == 0) rowsum[row] = s;
}

// ---------------------------------------------------------------------------
// Kernel 2 (1 block, 256 threads): logits + softmax + dy-grouped weights.
// logits[k][b] = (1/HW) * [ sum_c G[b,c]*Srange(b,c,dy_k) + (H-|dy|)*W*qb[b] ]
//   G[b,c]  = sum_d Qp[b,d]*Wk[d,c],  Qp[b,d] = sum_c Wq[d,c]*xmean[b,c]+bq[d]
//   Srange  = total rowsum minus |dy| head/tail rows (zero-padded row shift)
// ---------------------------------------------------------------------------
__global__ __launch_bounds__(256) void k2_logits(const float* __restrict__ rowsum,
                                                 const float* __restrict__ Wq,
                                                 const float* __restrict__ bq,
                                                 const float* __restrict__ Wk,
                                                 const float* __restrict__ bk,
                                                 const float* __restrict__ scaling,
                                                 float* __restrict__ wcoef,
                                                 DyInfo info) {
    __shared__ float tot[256];
    __shared__ float headc[256][4];
    __shared__ float tailc[256][4];
    __shared__ float xmean[256];
    __shared__ float Qp[256];
    __shared__ float G[256];
    __shared__ float qb[16];
    __shared__ float logit[8 * 16];

    int t = threadIdx.x;                 // t = b*16 + c  (or b*16 + d)
    {
        const float* base = rowsum + (size_t)t * Hh;
        float s = 0.f, h[4], tl[4];
        for (int r = 0; r < Hh; ++r) {
            float v = base[r];
            s += v;
            if (r < 4) h[r] = v;
            if (r >= Hh - 4) tl[r - (Hh - 4)] = v;
        }
        float hc = 0.f, tc = 0.f;
        #pragma unroll
        for (int k = 0; k < 4; ++k) { hc += h[k];      headc[t][k] = hc; }
        #pragma unroll
        for (int k = 0; k < 4; ++k) { tc += tl[3 - k]; tailc[t][k] = tc; }
        tot[t] = s;
        xmean[t] = s * (1.0f / (Hh * Ww));
    }
    __syncthreads();
    {   // Q_pooled
        int b = t >> 4, d = t & 15;
        float s = bq[d];
        for (int c = 0; c < Cc; ++c) s += Wq[d * Cc + c] * xmean[b * 16 + c];
        Qp[t] = s;
    }
    __syncthreads();
    {   // G and qb
        int b = t >> 4, c = t & 15;
        float s = 0.f;
        for (int d = 0; d < Cc; ++d) s += Qp[b * 16 + d] * Wk[d * Cc + c];
        G[t] = s;
    }
    if (t < 16) {
        float s = 0.f;
        for (int d = 0; d < Cc; ++d) s += Qp[t * 16 + d] * bk[d];
        qb[t] = s;
    }
    __syncthreads();
    if (t < 128) {                        // k = t>>4, b = t&15
        int k = t >> 4, b = t & 15;
        int dy = info.dy[k];
        float s = 0.f;
        for (int c = 0; c < Cc; ++c) {
            int idx = b * 16 + c;
            float sr = tot[idx];
            if (dy > 0)       sr -= tailc[idx][dy - 1];
            else if (dy < 0)  sr -= headc[idx][-dy - 1];
            s += G[idx] * sr;
        }
        s += ((float)(Hh - (dy < 0 ? -dy : dy))) * (float)Ww * qb[b];
        logit[k * 16 + b] = s * (1.0f / (Hh * Ww));
    }
    __syncthreads();
    if (t < 16) {                         // softmax per batch, group by dy
        int b = t;
        float mx = -3.0e38f;
        for (int k = 0; k < 8; ++k) mx = fmaxf(mx, logit[k * 16 + b]);
        float denom = fabsf(scaling[0]) + 1e-6f;
        float e[8], se = 0.f;
        for (int k = 0; k < 8; ++k) { e[k] = expf((logit[k * 16 + b] - mx) / denom); se += e[k]; }
        float wj[8];
        for (int j = 0; j < 8; ++j) wj[j] = 0.f;
        for (int k = 0; k < 8; ++k) wj[info.jmap[k]] += e[k] / se;
        for (int j = 0; j < 8; ++j) wcoef[b * 8 + j] = wj[j];
    }
}

// ---------------------------------------------------------------------------
// Kernel 3 (fused heavy pass): per block = (batch b, 8 output rows, 64-px strip).
// Stage 16 halo rows of A_send * M (1x1 conv via v_wmma_f32_16x16x4_f32, full f32)
// into LDS, then accumulate the dy-shifted weighted sum into float4 outputs.
// LDS: 16 rows * 16 ch * 64 px * 4B = 64 KiB (gfx1250 WGP has 320 KiB).
// The 3x3 max-pool is branchless: clamp coords (always-valid load) + VALU select.
// ---------------------------------------------------------------------------
__global__ __launch_bounds__(256) void k3_fused(const float* __restrict__ x,
                                                const float* __restrict__ Wm,
                                                const float* __restrict__ bm,
                                                const float* __restrict__ wcoef,
                                                float* __restrict__ out,
                                                DyInfo info) {
    __shared__ float lds[16 * 16 * 64];   // [row_local][out_ch][col]
    __shared__ float wsh[8];              // wcoef[b, 0..ndy)

    const int b    = blockIdx.z;
    const int y0   = blockIdx.y * 8;
    const int x0   = blockIdx.x * 64;
    const int t    = threadIdx.x;
    const int wave = t >> 5;
    const int lane = t & 31;
    const int n    = lane & 15;           // pixel/N (A: M, B/D: N)
    const int h2   = lane >> 4;           // lane half

    if (t < 8) wsh[t] = wcoef[b * 8 + t];

    // A operand = Wm, ISA layout: lane -> M (= out ch), VGPR v + 2*half -> K (= in ch)
    v2f a[4];
    #pragma unroll
    for (int ck = 0; ck < 4; ++ck) {
        a[ck].x = Wm[n * Cc + 4 * ck + 2 * h2];
        a[ck].y = Wm[n * Cc + 4 * ck + 2 * h2 + 1];
    }
    float bias[8];
    #pragma unroll
    for (int v = 0; v < 8; ++v) bias[v] = bm[v + 8 * h2];   // D layout: M = v + 8*half

    const float* x3 = x + ((size_t)(b * Cc + 3)) * (Hh * Ww);

    // ---- compute phase: 8 waves x 2 rows = 16 staged rows [y0-4, y0+11] ----
    for (int rr = 0; rr < 2; ++rr) {
        int r_local = wave * 2 + rr;
        int r = y0 - 4 + r_local;
        if (r >= 0 && r < Hh) {
            for (int tile = 0; tile < 4; ++tile) {
                int px = x0 + tile * 16 + n;

                // A_send: 3x3 max-pool (SAME, -inf pad) of x[:,3] then > 0.1.
                // Branchless: clamp coords so every load is valid, mask with select.
                float mx = -3.0e38f;
                #pragma unroll
                for (int yy = -1; yy <= 1; ++yy) {
                    int ry  = r + yy;
                    int ryc = min(max(ry, 0), Hh - 1);
                    bool rok = (ry == ryc);
                    #pragma unroll
                    for (int xx = -1; xx <= 1; ++xx) {
                        int cx  = px + xx;
                        int cxc = min(max(cx, 0), Ww - 1);
                        float v = x3[ryc * Ww + cxc];
                        bool ok = rok && (cx == cxc);
                        float cand = fmaxf(mx, v);
                        mx = ok ? cand : mx;
                    }
                }
                float as = (mx > 0.1f) ? 1.0f : 0.0f;

                // D(16xc,16px) = Wm(16x16) * X(16c,16px): 4 chained K=4 WMMAs
                v8f acc = {};
                #pragma unroll
                for (int ck = 0; ck < 4; ++ck) {
                    int c0 = 4 * ck + 2 * h2;   // B layout: lane -> N, v+2*half -> K
                    v2f bf;
                    bf.x = x[(((size_t)(b * Cc + c0    ) * Hh) + r) * Ww + px];
                    bf.y = x[(((size_t)(b * Cc + c0 + 1) * Hh) + r) * Ww + px];
                    acc = __builtin_amdgcn_wmma_f32_16x16x4_f32(
                        false, a[ck], false, bf, (short)0, acc, false, false);
                }
                int col = tile * 16 + n;
                #pragma unroll
                for (int v = 0; v < 8; ++v) {
                    int o = v + 8 * h2;
                    lds[(r_local * 16 + o) * 64 + col] = (acc[v] + bias[v]) * as;
                }
            }
        }
    }
    __syncthreads();

    // ---- accumulation phase: out[b,c,y,x] = sum_j w[b,j]*AM[y-dy_j] ----
    const float4* lds4 = (const float4*)lds;
    #pragma unroll
    for (int i = 0; i < 8; ++i) {
        int m  = t + 256 * i;             // 2048 float4 tasks
        int q  = m & 15;                  // float4 column within strip
        int yl = (m >> 4) & 7;
        int c  = m >> 7;
        int y  = y0 + yl;
        float4 acc = make_float4(0.f, 0.f, 0.f, 0.f);
        for (int j = 0; j < info.ndy; ++j) {
            int r = y - info.dylist[j];
            if (r < 0 || r >= Hh) continue;
            int rl = r - (y0 - 4);        // guaranteed in [0,15]
            float wv = wsh[j];
            float4 v = lds4[(rl * 16 + c) * 16 + q];
            acc.x += wv * v.x; acc.y += wv * v.y;
            acc.z += wv * v.z; acc.w += wv * v.w;
        }
        float4* op = (float4*)(out + (((size_t)(b * Cc + c) * Hh) + y) * Ww + x0 + 4 * q);
        *op = acc;
    }
}

// ---------------------------------------------------------------------------
// Host: CPython-compatible MT19937 + random.Random(0).sample(offsets, 8)
// ---------------------------------------------------------------------------
namespace pymt {
struct MT {
    uint32_t mt[624];
    int mti;
    void init_genrand(uint32_t s) {
        mt[0] = s;
        for (mti = 1; mti < 624; ++mti)
            mt[mti] = 1812433253u * (mt[mti - 1] ^ (mt[mti - 1] >> 30)) + (uint32_t)mti;
    }
    void init_by_array(const uint32_t* key, int klen) {
        init_genrand(19650218u);
        int i = 1, j = 0;
        int k = (624 > klen) ? 624 : klen;
        for (; k; --k) {
            mt[i] = (mt[i] ^ ((mt[i - 1] ^ (mt[i - 1] >> 30)) * 1664525u)) + key[j] + (uint32_t)j;
            ++i; ++j;
            if (i >= 624) { mt[0] = mt[623]; i = 1; }
            if (j >= klen) j = 0;
        }
        for (k = 623; k; --k) {
            mt[i] = (mt[i] ^ ((mt[i - 1] ^ (mt[i - 1] >> 30)) * 1566083941u)) - (uint32_t)i;
            ++i;
            if (i >= 624) { mt[0] = mt[623]; i = 1; }
        }
        mt[0] = 0x80000000u;
        mti = 624;
    }
    uint32_t genrand() {
        if (mti >= 624) {
            for (int kk = 0; kk < 624; ++kk) {
                uint32_t y = (mt[kk] & 0x80000000u) | (mt[(kk + 1) % 624] & 0x7fffffffu);
                mt[kk] = mt[(kk + 397) % 624] ^ (y >> 1) ^ ((y & 1u) ? 0x9908b0dfu : 0u);
            }
            mti = 0;
        }
        uint32_t y = mt[mti++];
        y ^= y >> 11;
        y ^= (y << 7)  & 0x9d2c5680u;
        y ^= (y << 15) & 0xefc60000u;
        y ^= y >> 18;
        return y;
    }
    uint32_t getrandbits(int k) { return genrand() >> (32 - k); }
    uint32_t randbelow(uint32_t nv) {
        int k = 32 - __builtin_clz(nv);
        uint32_t r = getrandbits(k);
        while (r >= nv) r = getrandbits(k);
        return r;
    }
};
} // namespace pymt

static void sample_offsets(int dys[8], int dxs[8]) {
    // _build_offsets(4): dy -4..4 outer, dx -4..4 inner, skip 3x3 center block
    int ody[72], odx[72], nn = 0;
    for (int dy = -4; dy <= 4; ++dy)
        for (int dx = -4; dx <= 4; ++dx) {
            if (dy >= -1 && dy <= 1 && dx >= -1 && dx <= 1) continue;
            ody[nn] = dy; odx[nn] = dx; ++nn;
        }
    // random.Random(0).sample(list72, 8): pool method (n=72 <= setsize=85)
    pymt::MT mt;
    uint32_t key = 0u;
    mt.init_by_array(&key, 1);
    int pool[72];
    for (int i = 0; i < 72; ++i) pool[i] = i;
    for (int i = 0; i < 8; ++i) {
        uint32_t j = mt.randbelow((uint32_t)(72 - i));
        int sel = pool[j];
        dys[i] = ody[sel]; dxs[i] = odx[sel];
        pool[j] = pool[72 - i - 1];
    }
}

extern "C" void kernel_launch(void* const* d_in, const int* in_sizes, int n_in,
                              void* d_out, int out_size, void* d_ws, size_t ws_size,
                              hipStream_t stream) {
    (void)in_sizes; (void)n_in; (void)out_size; (void)ws_size;
    const float* x       = (const float*)d_in[0];
    const float* Wq      = (const float*)d_in[1];
    const float* bq      = (const float*)d_in[2];
    const float* Wk      = (const float*)d_in[3];
    const float* bk      = (const float*)d_in[4];
    const float* Wm      = (const float*)d_in[5];
    const float* bm      = (const float*)d_in[6];
    const float* scaling = (const float*)d_in[7];
    float* out = (float*)d_out;

    float* rowsum = (float*)d_ws;         // 65536 floats
    float* wcoef  = rowsum + 65536;       // 16*8 floats

    int dys[8], dxs[8];
    sample_offsets(dys, dxs);             // dx is irrelevant (reference shift bug)

    DyInfo info;
    int ndy = 0;
    for (int k = 0; k < 8; ++k) {
        info.dy[k] = dys[k];
        int j = -1;
        for (int q = 0; q < ndy; ++q)
            if (info.dylist[q] == dys[k]) { j = q; break; }
        if (j < 0) { j = ndy; info.dylist[ndy++] = dys[k]; }
        info.jmap[k] = j;
    }
    for (int q = ndy; q < 8; ++q) info.dylist[q] = 0;
    info.ndy = ndy;

    k1_rowsum<<<8192, 256, 0, stream>>>(x, rowsum);
    k2_logits<<<1, 256, 0, stream>>>(rowsum, Wq, bq, Wk, bk, scaling, wcoef, info);
    dim3 g3(Ww / 64, Hh / 8, Bb);
    k3_fused<<<g3, 256, 0, stream>>>(x, Wm, bm, wcoef, out, info);
}